// WeightClassMagLoss_27814208209652
// MI455X (gfx1250) — compile-verified
//
#include <hip/hip_runtime.h>
#include <math.h>
#include <stdint.h>

// ---------------- tunables ----------------
#define BDIM 256                         // 8 waves (wave32)
#define TILE_F4 2048                     // float4s per tile = 8192 floats = 32 KB
#define TILE_FLOATS (TILE_F4 * 4)
#define NBUF 3                           // 3-deep async pipeline, 96 KB LDS
#define LOADS_PER_THREAD (TILE_F4 / BDIM)            // 8 b128 async loads / thread / tile
#define PIPE_OUTSTANDING ((NBUF - 1) * LOADS_PER_THREAD) // 16: tiles t+1,t+2 in flight

static constexpr float kScale   = 64.0f;
static constexpr float kUA      = 110.0f;
static constexpr float kLambdaG = 20.0f;
static constexpr float kLog2e   = 1.4426950408889634f;
static constexpr float kLn2     = 0.6931471805599453f;
static constexpr float kNegBig  = -3.0e38f;

// ---------------- CDNA5 helpers ----------------
#if __has_builtin(__builtin_amdgcn_s_wait_asynccnt)
#define WAIT_ASYNCCNT(n) __builtin_amdgcn_s_wait_asynccnt(n)
#else
#define WAIT_ASYNCCNT(n) asm volatile("s_wait_asynccnt %0" ::"i"(n) : "memory")
#endif

__device__ __forceinline__ void async_load_b128_gvs(uint32_t lds_byte_addr,
                                                    uint32_t vbyte_off,
                                                    const float* sbase) {
    // GLOBAL_LOAD_ASYNC_TO_LDS_B128, GVS mode: addr = SGPR64 base + VGPR32 byte offset.
    // vdst = LDS byte address (VGPR). Tracked with ASYNCcnt.
    asm volatile("global_load_async_to_lds_b128 %0, %1, %2"
                 :: "v"(lds_byte_addr), "v"(vbyte_off),
                    "s"((uint64_t)(uintptr_t)sbase)
                 : "memory");
}

__device__ __forceinline__ float fexp2(float x) {
#if __has_builtin(__builtin_amdgcn_exp2f)
    return __builtin_amdgcn_exp2f(x);      // v_exp_f32
#else
    return exp2f(x);
#endif
}
__device__ __forceinline__ float flog2(float x) {
#if __has_builtin(__builtin_amdgcn_logf)
    return __builtin_amdgcn_logf(x);       // v_log_f32
#else
    return log2f(x);
#endif
}

// Merge two online-logsumexp states (base-2 domain).
__device__ __forceinline__ void lse_merge(float& m, float& s, float mo, float so) {
    float mn = fmaxf(m, mo);
    s = s * fexp2(m - mn) + so * fexp2(mo - mn);
    m = mn;
}

// ---------------- kernel 1: one workgroup per row ----------------
__global__ __launch_bounds__(BDIM) void magface_row_kernel(
    const float* __restrict__ ct, const float* __restrict__ ctm,
    const float* __restrict__ xnorm, const float* __restrict__ alpha,
    const int* __restrict__ target, float* __restrict__ row_loss, int C)
{
    __shared__ __align__(16) float buf[NBUF * TILE_FLOATS];
    __shared__ float red_m[BDIM / 32];
    __shared__ float red_s[BDIM / 32];

    const int row = blockIdx.x;
    const int tid = threadIdx.x;
    const float* rowp = ct + (size_t)row * (size_t)C;
    const int c_f4 = C >> 2;                                   // C divisible by 4
    const int ntiles = (c_f4 + TILE_F4 - 1) / TILE_F4;

    // Low 32 bits of a generic LDS pointer == flat/DS LDS byte address.
    const uint32_t lds_base = (uint32_t)(uintptr_t)(void*)buf;
    const float K = kScale * kLog2e;   // logits scaled into base-2 exponent domain

    auto issue_tile = [&](int t) {
        const uint32_t boff = (uint32_t)(t % NBUF) * (uint32_t)(TILE_FLOATS * 4);
        const int base_f4 = t * TILE_F4;
#pragma unroll
        for (int k = 0; k < LOADS_PER_THREAD; ++k) {
            const int slot = k * BDIM + tid;
            int j = base_f4 + slot;
            j = (j < c_f4) ? j : (c_f4 - 1);      // clamp (never mask EXEC):
            async_load_b128_gvs(lds_base + boff + (uint32_t)slot * 16u,
                                (uint32_t)j * 16u, rowp);   // ASYNCcnt stays exact
        }
    };

    // Prime the pipeline.
    issue_tile(0);
    if (ntiles > 1) issue_tile(1);

    float m = kNegBig;                 // running max of z = 64*cos*log2e
    float s = 0.0f;                    // running sum of 2^(z - m)

    // ---- full tiles: no bounds guards ----
    for (int t = 0; t < ntiles - 1; ++t) {
        if (t + 2 < ntiles) issue_tile(t + 2);

        asm volatile("" ::: "memory");
        WAIT_ASYNCCNT(PIPE_OUTSTANDING);   // tile t done (in-order); t+1,t+2 in flight
        asm volatile("" ::: "memory");

        const float4* tb = reinterpret_cast<const float4*>(&buf[(t % NBUF) * TILE_FLOATS]);
        float4 v[LOADS_PER_THREAD];
#pragma unroll
        for (int k = 0; k < LOADS_PER_THREAD; ++k)
            v[k] = tb[k * BDIM + tid];                         // ds_load_b128

        float rmax = fmaxf(fmaxf(v[0].x, v[0].y), fmaxf(v[0].z, v[0].w));
#pragma unroll
        for (int k = 1; k < LOADS_PER_THREAD; ++k)
            rmax = fmaxf(rmax, fmaxf(fmaxf(v[k].x, v[k].y), fmaxf(v[k].z, v[k].w)));

        const float mn = fmaxf(m, rmax * K);
        float a0 = 0.0f, a1 = 0.0f, a2 = 0.0f, a3 = 0.0f;
#pragma unroll
        for (int k = 0; k < LOADS_PER_THREAD; ++k) {
            a0 += fexp2(fmaf(v[k].x, K, -mn));   // v_fma + v_exp per element
            a1 += fexp2(fmaf(v[k].y, K, -mn));
            a2 += fexp2(fmaf(v[k].z, K, -mn));
            a3 += fexp2(fmaf(v[k].w, K, -mn));
        }
        s = s * fexp2(m - mn) + ((a0 + a1) + (a2 + a3));
        m = mn;
    }

    // ---- tail tile: guarded ----
    {
        const int t = ntiles - 1;
        asm volatile("" ::: "memory");
        WAIT_ASYNCCNT(0);
        asm volatile("" ::: "memory");

        const float4* tb = reinterpret_cast<const float4*>(&buf[(t % NBUF) * TILE_FLOATS]);
        const int base_f4 = t * TILE_F4;

        float z[LOADS_PER_THREAD][4];
        float cmax = kNegBig;
#pragma unroll
        for (int k = 0; k < LOADS_PER_THREAD; ++k) {
            const int slot = k * BDIM + tid;
            const bool ok = (base_f4 + slot) < c_f4;
            const float4 v = tb[slot];
            z[k][0] = ok ? v.x * K : kNegBig;
            z[k][1] = ok ? v.y * K : kNegBig;
            z[k][2] = ok ? v.z * K : kNegBig;
            z[k][3] = ok ? v.w * K : kNegBig;
            cmax = fmaxf(cmax, fmaxf(fmaxf(z[k][0], z[k][1]), fmaxf(z[k][2], z[k][3])));
        }
        const float mn = fmaxf(m, cmax);
        float a0 = 0.0f, a1 = 0.0f, a2 = 0.0f, a3 = 0.0f;
#pragma unroll
        for (int k = 0; k < LOADS_PER_THREAD; ++k) {
            a0 += fexp2(z[k][0] - mn);
            a1 += fexp2(z[k][1] - mn);
            a2 += fexp2(z[k][2] - mn);
            a3 += fexp2(z[k][3] - mn);
        }
        s = s * fexp2(m - mn) + ((a0 + a1) + (a2 + a3));
        m = mn;
    }

    // ---- cross-lane merge (wave32) ----
#pragma unroll
    for (int off = 16; off >= 1; off >>= 1) {
        const float mo = __shfl_xor(m, off, 32);
        const float so = __shfl_xor(s, off, 32);
        lse_merge(m, s, mo, so);
    }
    const int wave = tid >> 5;
    if ((tid & 31) == 0) { red_m[wave] = m; red_s[wave] = s; }
    __syncthreads();

    if (tid == 0) {
        float M = red_m[0], S = red_s[0];
#pragma unroll
        for (int w = 1; w < BDIM / 32; ++w) lse_merge(M, S, red_m[w], red_s[w]);

        // Target-column correction: swap ct[row,tgt] -> ctm[row,tgt] in the sum.
        const int tg = target[row];
        const float z_ct  = rowp[tg] * K;
        const float z_ctm = ctm[(size_t)row * (size_t)C + tg] * K;
        const float mn = fmaxf(M, z_ctm);
        const float S2 = S * fexp2(M - mn) + fexp2(z_ctm - mn) - fexp2(z_ct - mn);

        // natural-log lse of corrected logits; target logit = z_ctm * ln2
        const float loss_id = kLn2 * (mn + flog2(S2) - z_ctm);

        const float xn = xnorm[row];
        const float loss_g = xn / (kUA * kUA) + 1.0f / xn;
        row_loss[row] = (loss_id + kLambdaG * loss_g) * alpha[tg];
    }
}

// ---------------- kernel 2: deterministic mean over B rows ----------------
__global__ __launch_bounds__(256) void mean_kernel(const float* __restrict__ rl,
                                                   float* __restrict__ out, int B)
{
    __shared__ float sm[8];
    float a = 0.0f;
    for (int i = threadIdx.x; i < B; i += 256) a += rl[i];
#pragma unroll
    for (int off = 16; off >= 1; off >>= 1) a += __shfl_xor(a, off, 32);
    if ((threadIdx.x & 31) == 0) sm[threadIdx.x >> 5] = a;
    __syncthreads();
    if (threadIdx.x == 0) {
        float t = 0.0f;
#pragma unroll
        for (int w = 0; w < 8; ++w) t += sm[w];
        out[0] = t / (float)B;
    }
}

// ---------------- launch ----------------
extern "C" void kernel_launch(void* const* d_in, const int* in_sizes, int n_in,
                              void* d_out, int out_size, void* d_ws, size_t ws_size,
                              hipStream_t stream) {
    const float* ct    = (const float*)d_in[0];   // [B, C]
    const float* ctm   = (const float*)d_in[1];   // [B, C]
    const float* xn    = (const float*)d_in[2];   // [B]
    const float* alpha = (const float*)d_in[3];   // [C]
    const int*   tgt   = (const int*)d_in[4];     // [B]

    const int B = in_sizes[2];
    const int C = in_sizes[3];

    float* row_loss = (float*)d_ws;               // B floats of scratch

    magface_row_kernel<<<B, BDIM, 0, stream>>>(ct, ctm, xn, alpha, tgt, row_loss, C);
    mean_kernel<<<1, 256, 0, stream>>>(row_loss, (float*)d_out, B);
}